// Model_18872086298796
// MI455X (gfx1250) — compile-verified
//
#include <hip/hip_runtime.h>
#include <math.h>

// xDeepFM forward for MI455X (gfx1250, wave32).
//  - single fused pass over x (106.6 MB -> HBM floor ~4.6us @ 23.3 TB/s)
//  - CIN as WMMA f32 GEMMs with the outer-product A operand generated during
//    LDS staging (Z tensor never materialized: saves ~700 MB of HBM traffic)
//  - 32x32 wave tiles: 4 v_wmma per k-step
//  - k-contiguous LDS tiles (pad 34): every WMMA fragment is one aligned
//    ds_load_b64 straight into the operand VGPR pair (no re-pairing movs)
//  - CIN H is a template constant so i=k/H, j=k%H lower to mul-shift
//  - next-K-tile weight prefetch via global_prefetch_b8

#define NUM_NUMERIC 13
#define NUM_CAT     26
#define CARD        1000
#define M_FIELDS    39
#define D_EMB       10
#define F_RAW       26013
#define ET_LD       40      // padded row stride of E_t
#define KP          34      // padded k stride of LDS tiles (even -> 8B align)

typedef __attribute__((ext_vector_type(2))) float v2f;
typedef __attribute__((ext_vector_type(8))) float v8f;

static __device__ __forceinline__ v8f wmma4(v2f a, v2f b, v8f c) {
  return __builtin_amdgcn_wmma_f32_16x16x4_f32(false, a, false, b,
                                               (short)0, c, false, false);
}

// ---------------- K1: fused linear + embedding (one block per batch row) ---
__global__ __launch_bounds__(256) void embed_linear_kernel(
    const float* __restrict__ x, const float* __restrict__ w_lin,
    const float* __restrict__ b_lin, const float* __restrict__ W_num,
    const float* __restrict__ W_cat,
    float* __restrict__ E_flat,   // (1024, 390)   [b][m*10+d]
    float* __restrict__ E_t,      // (10240, 40)   [(b*10+d)][m]
    float* __restrict__ lin_out)  // (1024)
{
  const int b    = blockIdx.x;
  const int t    = threadIdx.x;
  const int lane = t & 31;
  const int wave = t >> 5;
  const float* xb = x + (size_t)b * F_RAW;

  if (t < NUM_NUMERIC * D_EMB) {            // numeric embeddings
    int m = t / D_EMB, d = t - m * D_EMB;
    float v = xb[m] * W_num[t];
    E_flat[(size_t)b * (M_FIELDS * D_EMB) + t] = v;
    E_t[((size_t)b * D_EMB + d) * ET_LD + m]   = v;
  }

  float lin = 0.f;
  if (t < NUM_NUMERIC) lin = xb[t] * w_lin[t];

  // categorical fields: one wave per field (strided over 26 fields)
  for (int f = wave; f < NUM_CAT; f += 8) {
    const float* xf = xb + NUM_NUMERIC + f * CARD;
    const float* Wf = W_cat + (size_t)f * CARD * D_EMB;
    const float* wl = w_lin + NUM_NUMERIC + f * CARD;
    float acc[D_EMB];
#pragma unroll
    for (int d = 0; d < D_EMB; ++d) acc[d] = 0.f;
    for (int c = lane; c < CARD; c += 32) {  // coalesced 128B wave reads
      float xv = xf[c];
      lin += xv * wl[c];
#pragma unroll
      for (int d = 0; d < D_EMB; ++d) acc[d] += xv * Wf[c * D_EMB + d];
    }
#pragma unroll
    for (int d = 0; d < D_EMB; ++d) {
#pragma unroll
      for (int off = 16; off > 0; off >>= 1)
        acc[d] += __shfl_xor(acc[d], off, 32);
    }
    if (lane == 0) {
      int m = NUM_NUMERIC + f;
#pragma unroll
      for (int d = 0; d < D_EMB; ++d) {
        E_flat[(size_t)b * (M_FIELDS * D_EMB) + m * D_EMB + d] = acc[d];
        E_t[((size_t)b * D_EMB + d) * ET_LD + m]               = acc[d];
      }
    }
  }

#pragma unroll
  for (int off = 16; off > 0; off >>= 1) lin += __shfl_xor(lin, off, 32);
  __shared__ float red[8];
  if (lane == 0) red[wave] = lin;
  __syncthreads();
  if (t == 0) {
    float s = b_lin[0];
    for (int w = 0; w < 8; ++w) s += red[w];
    lin_out[b] = s;
  }
}

// ---------------- generic WMMA f32 GEMM: C = A(MxK)*B(KxN) + bias [, relu] -
// 256 threads = 8 waves; block tile 256x32; wave tile 32x32 (4 accumulators);
// K staged in 32-deep LDS tiles, stored k-contiguous for b64 fragment loads.
// M must be a multiple of 256.
__global__ __launch_bounds__(256) void gemm_wmma_kernel(
    const float* __restrict__ A, const float* __restrict__ Bm,
    const float* __restrict__ bias, float* __restrict__ C,
    int Ndim, int Kdim, int lda, int ldb, int ldc, int relu)
{
  __shared__ float As[256][KP];  // [m][k]
  __shared__ float Bs[32][KP];   // [n][k]
  const int t = threadIdx.x;
  const int lane = t & 31, wave = t >> 5;
  const int lh = lane & 15, hh = lane >> 4;
  const int row0 = blockIdx.y * 256;
  const int col0 = blockIdx.x * 32;
  const int mw = wave * 32;      // wave's row offset inside the block tile

  v8f c00 = {}, c01 = {}, c10 = {}, c11 = {};
  for (int k0 = 0; k0 < Kdim; k0 += 32) {
#pragma unroll 4
    for (int i = 0; i < 32; ++i) {           // stage A 256x32 (coalesced in k)
      int e = t + i * 256;
      int k = e & 31, r = e >> 5;
      int gk = k0 + k;
      As[r][k] = (gk < Kdim) ? A[(size_t)(row0 + r) * lda + gk] : 0.f;
    }
#pragma unroll
    for (int i = 0; i < 4; ++i) {            // stage B 32x32 (coalesced in n)
      int e = t + i * 256;
      int n = e & 31, k = e >> 5;
      int gk = k0 + k, gn = col0 + n;
      Bs[n][k] = (gk < Kdim && gn < Ndim) ? Bm[(size_t)gk * ldb + gn] : 0.f;
    }
    if (k0 + 32 + (t >> 3) < Kdim)           // prefetch next B K-tile
      __builtin_prefetch(&Bm[(size_t)(k0 + 32 + (t >> 3)) * ldb + col0], 0, 1);
    __syncthreads();
#pragma unroll
    for (int kk = 0; kk < 32; kk += 4) {
      // ISA 7.12.2 f32 frags: v0 = K(0|2), v1 = K(1|3) by lane half
      int kb = kk + 2 * hh;                  // even -> 8B-aligned b64 loads
      v2f a0 = *(const v2f*)&As[mw + lh][kb];
      v2f a1 = *(const v2f*)&As[mw + 16 + lh][kb];
      v2f b0 = *(const v2f*)&Bs[lh][kb];
      v2f b1 = *(const v2f*)&Bs[16 + lh][kb];
      c00 = wmma4(a0, b0, c00);  c01 = wmma4(a0, b1, c01);
      c10 = wmma4(a1, b0, c10);  c11 = wmma4(a1, b1, c11);
    }
    __syncthreads();
  }
  const int gc0 = col0 + lh, gc1 = col0 + 16 + lh;
  const float bv0 = (gc0 < Ndim && bias) ? bias[gc0] : 0.f;
  const float bv1 = (gc1 < Ndim && bias) ? bias[gc1] : 0.f;
#pragma unroll
  for (int v = 0; v < 8; ++v) {              // C: VGPR v -> M = v + 8*laneHalf
    int r0 = row0 + mw + v + 8 * hh;
    int r1 = r0 + 16;
    float o00 = c00[v] + bv0, o10 = c10[v] + bv0;
    float o01 = c01[v] + bv1, o11 = c11[v] + bv1;
    if (relu) { o00 = fmaxf(o00, 0.f); o10 = fmaxf(o10, 0.f);
                o01 = fmaxf(o01, 0.f); o11 = fmaxf(o11, 0.f); }
    if (gc0 < Ndim) { C[(size_t)r0 * ldc + gc0] = o00;
                      C[(size_t)r1 * ldc + gc0] = o10; }
    if (gc1 < Ndim) { C[(size_t)r0 * ldc + gc1] = o01;
                      C[(size_t)r1 * ldc + gc1] = o11; }
  }
}

// ---------------- CIN WMMA GEMM with fused outer-product A-operand ---------
// rows = (b,d) pairs (10240). A[row][i*H+j] = E_t[row][i]*Curr[row][j],
// generated during LDS staging. H/CLD are compile-time so the k -> (i,j)
// decomposition lowers to mul-shift (no VALU integer division).
template <int H, int CLD>
__global__ __launch_bounds__(256) void cin_gemm_wmma_kernel(
    const float* __restrict__ Et, const float* __restrict__ Curr,
    const float* __restrict__ W, const float* __restrict__ bias,
    float* __restrict__ C, int Kdim, int Ndim)
{
  __shared__ float As[256][KP];
  __shared__ float Bs[32][KP];
  const int t = threadIdx.x;
  const int lane = t & 31, wave = t >> 5;
  const int lh = lane & 15, hh = lane >> 4;
  const int row0 = blockIdx.y * 256;
  const int col0 = blockIdx.x * 32;
  const int mw = wave * 32;

  v8f c00 = {}, c01 = {}, c10 = {}, c11 = {};
  for (int k0 = 0; k0 < Kdim; k0 += 32) {
#pragma unroll 4
    for (int i = 0; i < 32; ++i) {           // fused outer-product staging
      int e = t + i * 256;
      int k = e & 31, r = e >> 5;
      int gk = k0 + k;
      float av = 0.f;
      if (gk < Kdim) {
        int ii = gk / H, jj = gk - ii * H;   // constant H -> mul-shift
        av = Et[(size_t)(row0 + r) * ET_LD + ii] *
             Curr[(size_t)(row0 + r) * CLD + jj];
      }
      As[r][k] = av;
    }
#pragma unroll
    for (int i = 0; i < 4; ++i) {
      int e = t + i * 256;
      int n = e & 31, k = e >> 5;
      int gk = k0 + k, gn = col0 + n;
      Bs[n][k] = (gk < Kdim && gn < Ndim) ? W[(size_t)gk * Ndim + gn] : 0.f;
    }
    if (k0 + 32 + (t >> 3) < Kdim)           // prefetch next W K-tile
      __builtin_prefetch(&W[(size_t)(k0 + 32 + (t >> 3)) * Ndim + col0], 0, 1);
    __syncthreads();
#pragma unroll
    for (int kk = 0; kk < 32; kk += 4) {
      int kb = kk + 2 * hh;
      v2f a0 = *(const v2f*)&As[mw + lh][kb];
      v2f a1 = *(const v2f*)&As[mw + 16 + lh][kb];
      v2f b0 = *(const v2f*)&Bs[lh][kb];
      v2f b1 = *(const v2f*)&Bs[16 + lh][kb];
      c00 = wmma4(a0, b0, c00);  c01 = wmma4(a0, b1, c01);
      c10 = wmma4(a1, b0, c10);  c11 = wmma4(a1, b1, c11);
    }
    __syncthreads();
  }
  const int gc0 = col0 + lh, gc1 = col0 + 16 + lh;
  const float bv0 = (gc0 < Ndim) ? bias[gc0] : 0.f;
  const float bv1 = (gc1 < Ndim) ? bias[gc1] : 0.f;
#pragma unroll
  for (int v = 0; v < 8; ++v) {
    int r0 = row0 + mw + v + 8 * hh;
    int r1 = r0 + 16;
    if (gc0 < Ndim) { C[(size_t)r0 * Ndim + gc0] = c00[v] + bv0;
                      C[(size_t)r1 * Ndim + gc0] = c10[v] + bv0; }
    if (gc1 < Ndim) { C[(size_t)r0 * Ndim + gc1] = c01[v] + bv1;
                      C[(size_t)r1 * Ndim + gc1] = c11[v] + bv1; }
  }
}

// ---------------- final fusion: DNN logit + CIN sum-pool logit + sigmoid ---
__global__ __launch_bounds__(256) void final_kernel(
    const float* __restrict__ h1, const float* __restrict__ dnn_lin_w,
    const float* __restrict__ dnn_lin_b,
    const float* __restrict__ c0, const float* __restrict__ c1,
    const float* __restrict__ c2,
    const float* __restrict__ cin_lin_w, const float* __restrict__ cin_lin_b,
    const float* __restrict__ lin_logit, const float* __restrict__ pred_b,
    float* __restrict__ out)
{
  const int b = blockIdx.x, t = threadIdx.x;
  const int lane = t & 31, wave = t >> 5;
  float acc = 0.f;
  for (int i = t; i < 400; i += 256)
    acc += h1[(size_t)b * 400 + i] * dnn_lin_w[i];
  const float* bufs[3] = {c0, c1, c2};
  for (int l = 0; l < 3; ++l) {
    const float* p = bufs[l] + (size_t)b * D_EMB * 200;
    for (int h = t; h < 200; h += 256) {
      float s = 0.f;
#pragma unroll
      for (int d = 0; d < D_EMB; ++d) s += p[d * 200 + h];  // sum over D
      acc += s * cin_lin_w[l * 200 + h];
    }
  }
#pragma unroll
  for (int off = 16; off > 0; off >>= 1) acc += __shfl_xor(acc, off, 32);
  __shared__ float red[8];
  if (lane == 0) red[wave] = acc;
  __syncthreads();
  if (t == 0) {
    float z = dnn_lin_b[0] + cin_lin_b[0] + lin_logit[b] + pred_b[0];
    for (int w = 0; w < 8; ++w) z += red[w];
    out[b] = 1.f / (1.f + expf(-z));
  }
}

// ---------------------------------------------------------------------------
extern "C" void kernel_launch(void* const* d_in, const int* in_sizes, int n_in,
                              void* d_out, int out_size, void* d_ws,
                              size_t ws_size, hipStream_t stream)
{
  (void)in_sizes; (void)n_in; (void)out_size; (void)ws_size;
  const float* x         = (const float*)d_in[0];
  const float* w_lin     = (const float*)d_in[1];
  const float* b_lin     = (const float*)d_in[2];
  const float* W_num     = (const float*)d_in[3];
  const float* W_cat     = (const float*)d_in[4];
  const float* cin_w0    = (const float*)d_in[5];
  const float* cin_b0    = (const float*)d_in[6];
  const float* cin_w1    = (const float*)d_in[7];
  const float* cin_b1    = (const float*)d_in[8];
  const float* cin_w2    = (const float*)d_in[9];
  const float* cin_b2    = (const float*)d_in[10];
  const float* cin_lin_w = (const float*)d_in[11];
  const float* cin_lin_b = (const float*)d_in[12];
  const float* dnn_w0    = (const float*)d_in[13];
  const float* dnn_b0    = (const float*)d_in[14];
  const float* dnn_w1    = (const float*)d_in[15];
  const float* dnn_b1    = (const float*)d_in[16];
  const float* dnn_lin_w = (const float*)d_in[17];
  const float* dnn_lin_b = (const float*)d_in[18];
  const float* pred_b    = (const float*)d_in[19];
  float* out = (float*)d_out;

  // workspace layout (floats), total ~29.7 MB
  float* ws     = (float*)d_ws;
  float* E_flat = ws;                       // 1024*390
  float* E_t    = E_flat + 1024 * 390;      // 10240*40 (padded)
  float* lin    = E_t + 10240 * ET_LD;      // 1024
  float* h0     = lin + 1024;               // 1024*400
  float* h1     = h0 + 1024 * 400;          // 1024*400
  float* c0     = h1 + 1024 * 400;          // 10240*200
  float* c1     = c0 + 10240 * 200;         // 10240*200
  float* c2     = c1 + 10240 * 200;         // 10240*200

  // 1) single pass over x: linear logit + all embeddings
  embed_linear_kernel<<<1024, 256, 0, stream>>>(x, w_lin, b_lin, W_num, W_cat,
                                                E_flat, E_t, lin);

  // 2) DNN branch: (1024x390)x(390x400) relu, (1024x400)x(400x400) relu
  gemm_wmma_kernel<<<dim3(13, 4), 256, 0, stream>>>(
      E_flat, dnn_w0, dnn_b0, h0, 400, 390, 390, 400, 400, 1);
  gemm_wmma_kernel<<<dim3(13, 4), 256, 0, stream>>>(
      h0, dnn_w1, dnn_b1, h1, 400, 400, 400, 400, 400, 1);

  // 3) CIN: fused outer-product GEMMs over rows = (b,d) pairs (10240)
  cin_gemm_wmma_kernel<39, 40><<<dim3(7, 40), 256, 0, stream>>>(
      E_t, E_t, cin_w0, cin_b0, c0, 39 * 39, 200);
  cin_gemm_wmma_kernel<200, 200><<<dim3(7, 40), 256, 0, stream>>>(
      E_t, c0, cin_w1, cin_b1, c1, 39 * 200, 200);
  cin_gemm_wmma_kernel<200, 200><<<dim3(7, 40), 256, 0, stream>>>(
      E_t, c1, cin_w2, cin_b2, c2, 39 * 200, 200);

  // 4) logits + sigmoid
  final_kernel<<<1024, 256, 0, stream>>>(h1, dnn_lin_w, dnn_lin_b, c0, c1, c2,
                                         cin_lin_w, cin_lin_b, lin, pred_b,
                                         out);
}